// ESN_87531433492838
// MI455X (gfx1250) — compile-verified
//
#include <hip/hip_runtime.h>
#include <hip/hip_bf16.h>
#include <math.h>

// Problem constants (from reference)
#define N_U      128
#define N_Y      64
#define N_XR     4096
#define T_STEPS  4096
#define ALPHA_F        0.9f
#define ONE_M_ALPHA_F  0.1f

// ELL sparse format for W (density 0.05 -> row nnz ~205 +/- 14; 320 is >8 sigma)
#define ELL_W    320

// Persistent step kernel shape: 128 blocks * 256 threads, 8 waves * 4 rows = 32 rows/block
#define STEP_BLOCKS 128
#define STEP_TPB    256

typedef __attribute__((ext_vector_type(2))) float v2f;
typedef __attribute__((ext_vector_type(8))) float v8f;

__device__ __forceinline__ v8f wmma_f32_16x16x4(v2f a, v2f b, v8f c) {
  // D = A(16x4 f32) x B(4x16 f32) + C(16x16 f32), full f32 precision
  return __builtin_amdgcn_wmma_f32_16x16x4_f32(
      /*neg_a=*/false, a, /*neg_b=*/false, b,
      /*c_mod=*/(short)0, c, /*reuse_a=*/false, /*reuse_b=*/false);
}

// ---------------------------------------------------------------------------
// Init: X row 0 = x0, barrier counter = 0 (re-done every call: graph-safe)
// ---------------------------------------------------------------------------
__global__ void esn_init(const float* __restrict__ x0, float* __restrict__ X,
                         unsigned* __restrict__ cnt) {
  const int i = blockIdx.x * blockDim.x + threadIdx.x;
  if (i < N_XR) X[i] = x0[i];
  if (i == 0) *cnt = 0u;
}

// ---------------------------------------------------------------------------
// Pack W (5% dense) into fixed-width ELL: one wave per row, wave32 stream
// compaction via ballot/popc. Entries beyond ELL_W dropped (prob ~0), tail
// padded with (0.0, idx 0) which contributes nothing to the dot product.
// ---------------------------------------------------------------------------
__global__ void __launch_bounds__(256)
esn_pack_ell(const float* __restrict__ W, float* __restrict__ vals,
             unsigned short* __restrict__ idxs) {
  const int lane = threadIdx.x & 31;
  const int wave = threadIdx.x >> 5;
  const int row  = blockIdx.x * 8 + wave;              // 4096 rows / 512 blocks
  const float* wr = W + (size_t)row * N_XR;
  float*          vr = vals + (size_t)row * ELL_W;
  unsigned short* ir = idxs + (size_t)row * ELL_W;

  int count = 0;
  for (int k0 = 0; k0 < N_XR; k0 += 32) {
    const float w = wr[k0 + lane];
    const bool  nz = (w != 0.0f);
    const unsigned m = (unsigned)__ballot(nz);
    const int pos = count + __popc(m & ((1u << lane) - 1u));
    if (nz && pos < ELL_W) { vr[pos] = w; ir[pos] = (unsigned short)(k0 + lane); }
    count += __popc(m);
  }
  if (count > ELL_W) count = ELL_W;
  for (int p = count + lane; p < ELL_W; p += 32) { vr[p] = 0.0f; ir[p] = 0; }
}

// ---------------------------------------------------------------------------
// Phase 1: B[T, N_XR] = UT[T, N_U] @ W_in^T   (f32 WMMA, one 16x16 tile/wave)
// ---------------------------------------------------------------------------
__global__ void __launch_bounds__(256)
esn_win_gemm(const float* __restrict__ UT, const float* __restrict__ Win,
             float* __restrict__ B) {
  const int lane = threadIdx.x & 31;
  const int wave = threadIdx.x >> 5;
  const int tile = blockIdx.x * 8 + wave;          // (T/16)*(N_XR/16) tiles
  const int tj   = tile & (N_XR / 16 - 1);         // 0..255
  const int tt   = tile >> 8;                      // 0..255
  const int l15  = lane & 15;
  const int hi   = lane >> 4;

  const float* Arow = UT  + (size_t)(tt * 16 + l15) * N_U;
  const float* Brow = Win + (size_t)(tj * 16 + l15) * N_U;
  v8f c = {};
#pragma unroll 4
  for (int k = 0; k < N_U; k += 4) {
    const int kk = k + 2 * hi;                     // ISA A/B lane-half K split
    v2f a = { Arow[kk], Arow[kk + 1] };
    v2f b = { Brow[kk], Brow[kk + 1] };
    c = wmma_f32_16x16x4(a, b, c);
  }
  float* out = B + (size_t)(tt * 16 + 8 * hi) * N_XR + tj * 16 + l15;
#pragma unroll
  for (int i = 0; i < 8; ++i) out[(size_t)i * N_XR] = c[i];
}

// ---------------------------------------------------------------------------
// Phase 2: persistent sequential reservoir update over ELL-sparse W.
// X[t+1] = 0.1*X[t] + 0.9*tanh(B[t] + W @ X[t]); device-wide barrier per step.
// x_prev staged into LDS via gfx1250 async-copy path; sparse gathers hit LDS.
// ---------------------------------------------------------------------------
__global__ void __launch_bounds__(STEP_TPB)
esn_steps(const float* __restrict__ vals, const unsigned short* __restrict__ idxs,
          const float* __restrict__ B, float* __restrict__ X,
          unsigned* __restrict__ cnt) {
  __shared__ float xs[N_XR];                       // 16 KB
  const int tid  = threadIdx.x;
  const int lane = tid & 31;
  const int wave = tid >> 5;
  const int rowBase = blockIdx.x * 32 + wave * 4;  // 4 rows per wave

  const float*          v0 = vals + (size_t)(rowBase + 0) * ELL_W;
  const float*          v1 = vals + (size_t)(rowBase + 1) * ELL_W;
  const float*          v2 = vals + (size_t)(rowBase + 2) * ELL_W;
  const float*          v3 = vals + (size_t)(rowBase + 3) * ELL_W;
  const unsigned short* i0 = idxs + (size_t)(rowBase + 0) * ELL_W;
  const unsigned short* i1 = idxs + (size_t)(rowBase + 1) * ELL_W;
  const unsigned short* i2 = idxs + (size_t)(rowBase + 2) * ELL_W;
  const unsigned short* i3 = idxs + (size_t)(rowBase + 3) * ELL_W;

  // LDS byte address of xs (generic LDS pointer low 32 bits = LDS offset)
  const unsigned xs_base = (unsigned)(uintptr_t)(&xs[0]);

  for (int t = 0; t < T_STEPS; ++t) {
    // ---- stage x_prev into LDS via async global->LDS copy (ASYNCcnt) ----
    const unsigned long long xpa = (unsigned long long)(X + (size_t)t * N_XR);
#pragma unroll
    for (int i = tid; i < N_XR / 4; i += STEP_TPB) {
      const unsigned goff  = (unsigned)(i * 16);
      const unsigned laddr = xs_base + goff;
      asm volatile("global_load_async_to_lds_b128 %0, %1, %2"
                   :: "v"(laddr), "v"(goff), "s"(xpa) : "memory");
    }
    asm volatile("s_wait_asynccnt 0x0" ::: "memory");
    __syncthreads();

    // ---- sparse dot products: 4 rows per wave, ELL_W/32 entries per lane ----
    float a0 = 0.f, a1 = 0.f, a2 = 0.f, a3 = 0.f;
#pragma unroll
    for (int j = 0; j < ELL_W / 32; ++j) {
      const int o = j * 32 + lane;
      a0 += v0[o] * xs[i0[o]];
      a1 += v1[o] * xs[i1[o]];
      a2 += v2[o] * xs[i2[o]];
      a3 += v3[o] * xs[i3[o]];
    }
    // wave32 butterfly reductions
#pragma unroll
    for (int off = 16; off > 0; off >>= 1) {
      a0 += __shfl_xor(a0, off, 32);
      a1 += __shfl_xor(a1, off, 32);
      a2 += __shfl_xor(a2, off, 32);
      a3 += __shfl_xor(a3, off, 32);
    }
    if (lane == 0) {
      const float acc[4] = { a0, a1, a2, a3 };
      float* xn = X + (size_t)(t + 1) * N_XR;
      const float* bt = B + (size_t)t * N_XR;
#pragma unroll
      for (int r = 0; r < 4; ++r) {
        const int row = rowBase + r;
        const float pre = bt[row] + acc[r];
        xn[row] = ONE_M_ALPHA_F * xs[row] + ALPHA_F * tanhf(pre);
      }
    }
    // ---- device-wide barrier (monotonic counter, thread-0 arrival/spin) ----
    __threadfence();
    __syncthreads();
    if (tid == 0) {
      __hip_atomic_fetch_add(cnt, 1u, __ATOMIC_RELEASE, __HIP_MEMORY_SCOPE_AGENT);
      const unsigned target = (unsigned)(t + 1) * STEP_BLOCKS;
      while (__hip_atomic_load(cnt, __ATOMIC_ACQUIRE, __HIP_MEMORY_SCOPE_AGENT) < target)
        __builtin_amdgcn_s_sleep(2);
    }
    __syncthreads();
  }
}

// ---------------------------------------------------------------------------
// Phase 3: Y[T, N_Y] = X_states[T, N_XR] @ Wout^T   (f32 WMMA)
// ---------------------------------------------------------------------------
__global__ void __launch_bounds__(256)
esn_readout_gemm(const float* __restrict__ Xs, const float* __restrict__ Wout,
                 float* __restrict__ Y) {
  const int lane = threadIdx.x & 31;
  const int wave = threadIdx.x >> 5;
  const int tile = blockIdx.x * 8 + wave;          // (T/16)*(N_Y/16) = 1024 tiles
  const int tj   = tile & 3;                       // 0..3
  const int tt   = tile >> 2;                      // 0..255
  const int l15  = lane & 15;
  const int hi   = lane >> 4;

  const float* Arow = Xs   + (size_t)(tt * 16 + l15) * N_XR;
  const float* Brow = Wout + (size_t)(tj * 16 + l15) * N_XR;
  v8f c = {};
#pragma unroll 8
  for (int k = 0; k < N_XR; k += 4) {
    const int kk = k + 2 * hi;
    v2f a = { Arow[kk], Arow[kk + 1] };
    v2f b = { Brow[kk], Brow[kk + 1] };
    c = wmma_f32_16x16x4(a, b, c);
  }
  float* out = Y + (size_t)(tt * 16 + 8 * hi) * N_Y + tj * 16 + l15;
#pragma unroll
  for (int i = 0; i < 8; ++i) out[(size_t)i * N_Y] = c[i];
}

// ---------------------------------------------------------------------------
extern "C" void kernel_launch(void* const* d_in, const int* in_sizes, int n_in,
                              void* d_out, int out_size, void* d_ws, size_t ws_size,
                              hipStream_t stream) {
  (void)in_sizes; (void)n_in; (void)out_size; (void)ws_size;
  const float* UT   = (const float*)d_in[0];   // [T, N_U]
  const float* x0   = (const float*)d_in[1];   // [N_XR]
  const float* Win  = (const float*)d_in[2];   // [N_XR, N_U]
  const float* W    = (const float*)d_in[3];   // [N_XR, N_XR]
  const float* Wout = (const float*)d_in[4];   // [N_Y, N_XR]
  float* Y = (float*)d_out;                    // [T*N_Y]

  char* ws = (char*)d_ws;
  const size_t offB = 0;                                               // 64 MB
  const size_t offX = offB + (size_t)T_STEPS * N_XR * sizeof(float);   // 64 MB + 16 KB
  const size_t offC = offX + (size_t)(T_STEPS + 1) * N_XR * sizeof(float);
  const size_t offV = offC + 1024;                                     // 5.24 MB
  const size_t offI = offV + (size_t)N_XR * ELL_W * sizeof(float);     // 2.62 MB

  float*          Bws  = (float*)(ws + offB);
  float*          Xws  = (float*)(ws + offX);
  unsigned*       cnt  = (unsigned*)(ws + offC);
  float*          vals = (float*)(ws + offV);
  unsigned short* idxs = (unsigned short*)(ws + offI);

  hipLaunchKernelGGL(esn_init, dim3(16), dim3(256), 0, stream, x0, Xws, cnt);
  hipLaunchKernelGGL(esn_pack_ell, dim3(N_XR / 8), dim3(256), 0, stream,
                     W, vals, idxs);
  hipLaunchKernelGGL(esn_win_gemm,
                     dim3((T_STEPS / 16) * (N_XR / 16) / 8), dim3(256), 0, stream,
                     UT, Win, Bws);
  hipLaunchKernelGGL(esn_steps, dim3(STEP_BLOCKS), dim3(STEP_TPB), 0, stream,
                     vals, idxs, Bws, Xws, cnt);
  hipLaunchKernelGGL(esn_readout_gemm,
                     dim3((T_STEPS / 16) * (N_Y / 16) / 8), dim3(256), 0, stream,
                     Xws + N_XR, Wout, Y);
}